// RSEncoderLayer_40888088658571
// MI455X (gfx1250) — compile-verified
//
#include <hip/hip_runtime.h>
#include <math.h>

#define NNODES 100000
#define NEDGES 3200000
#define FDIM   128
#define TREP   4
#define EPSV   1e-15f

typedef __attribute__((ext_vector_type(2))) float v2f;
typedef __attribute__((ext_vector_type(8))) float v8f;

// ---------------- degree / dinv ----------------
__global__ void deg_init_kernel(float* __restrict__ deg) {
    int i = blockIdx.x * blockDim.x + threadIdx.x;
    if (i < NNODES) deg[i] = 1.0f;              // self-loop
}

__global__ void deg_accum_kernel(const int* __restrict__ dst, float* __restrict__ deg) {
    int e = blockIdx.x * blockDim.x + threadIdx.x;
    if (e < NEDGES) atomicAdd(&deg[dst[e]], 1.0f);
}

__global__ void dinv_kernel(float* __restrict__ deg) {
    int i = blockIdx.x * blockDim.x + threadIdx.x;
    if (i < NNODES) deg[i] = rsqrtf(deg[i]);    // deg becomes dinv in place
}

// ---------------- zero accumulator (t=0 slice of d_out) ----------------
__global__ void zero_acc_kernel(float4* __restrict__ acc4) {
    int i = blockIdx.x * blockDim.x + threadIdx.x;
    if (i < NNODES * FDIM / 4) acc4[i] = make_float4(0.f, 0.f, 0.f, 0.f);
}

// ---------------- GEMM: hs = dinv[row] * (x @ W), via V_WMMA_F32_16X16X4_F32 ----
// Block = 256 threads = 8 waves. Block handles one 16-row M tile; wave w handles
// N-columns [w*16, w*16+16). K=128 -> 32 WMMA(K=4) issues per tile.
__global__ void __launch_bounds__(256)
gemm_scale_kernel(const float* __restrict__ x, const float* __restrict__ W,
                  const float* __restrict__ dinv, float* __restrict__ hs) {
    __shared__ float xs[16 * 132];              // padded stride: stride 132 -> conflict-free
    const int tile = blockIdx.x;
    const int row0 = tile * 16;
    const int tid  = threadIdx.x;
    const int wave = tid >> 5;                  // 0..7  (N-column tile)
    const int lane = tid & 31;

    // cooperative stage of the 16x128 x-tile into LDS
    for (int i = tid; i < 16 * 128; i += 256) {
        int r = i >> 7, c = i & 127;
        int gr = row0 + r;
        xs[r * 132 + c] = (gr < NNODES) ? x[gr * FDIM + c] : 0.0f;
    }
    __syncthreads();

    const int half = lane >> 4;                 // 0: K pair {0,1} | 1: K pair {2,3}
    const int l16  = lane & 15;
    const int col  = wave * 16 + l16;

    v8f acc = {};
    for (int kk = 0; kk < 32; ++kk) {
        const int kbase = kk * 4 + half * 2;
        // A fragment (16x4): lanes 0-15 M=l16 K={kb,kb+1}; lanes 16-31 K={kb,kb+1}+2
        v2f a;
        a.x = xs[l16 * 132 + kbase + 0];
        a.y = xs[l16 * 132 + kbase + 1];
        // B fragment (4x16): VGPR0 = row kbase, VGPR1 = row kbase+1, N = col
        v2f b;
        b.x = W[(kbase + 0) * FDIM + col];
        b.y = W[(kbase + 1) * FDIM + col];
        acc = __builtin_amdgcn_wmma_f32_16x16x4_f32(
            /*neg_a=*/false, a, /*neg_b=*/false, b,
            /*c_mod=*/(short)0, acc, /*reuse_a=*/false, /*reuse_b=*/false);
    }

    // C/D layout: VGPR r -> M = r + half*8, N = col. Scale row by dinv on store.
    for (int r = 0; r < 8; ++r) {
        int grow = row0 + half * 8 + r;
        if (grow < NNODES) hs[(size_t)grow * FDIM + col] = acc[r] * dinv[grow];
    }
}

// ---------------- edge scatter: acc[dst] += hs[src], one wave per edge -------
__global__ void __launch_bounds__(256)
scatter_kernel(const int* __restrict__ src, const int* __restrict__ dst,
               const float* __restrict__ hs, float* __restrict__ acc) {
    int e = blockIdx.x * 8 + (threadIdx.x >> 5);
    if (e >= NEDGES) return;
    int lane = threadIdx.x & 31;
    int s = src[e], d = dst[e];
    float4 v = ((const float4*)(hs + (size_t)s * FDIM))[lane];   // 512B coalesced (L2)
    float* p = acc + (size_t)d * FDIM + lane * 4;
    atomicAdd(p + 0, v.x);
    atomicAdd(p + 1, v.y);
    atomicAdd(p + 2, v.z);
    atomicAdd(p + 3, v.w);
}

// ---------------- finalize: out = dinv*(acc+hs)+b; write T copies + expmap0 ---
__global__ void __launch_bounds__(256)
finalize_kernel(const float* __restrict__ hs, const float* __restrict__ dinv,
                const float* __restrict__ bias, float* __restrict__ out) {
    int row = blockIdx.x * 8 + (threadIdx.x >> 5);
    if (row >= NNODES) return;
    int lane = threadIdx.x & 31;
    float di = dinv[row];
    size_t base = (size_t)row * FDIM;

    float4 a  = ((const float4*)(out + base))[lane];   // accumulator (t=0 slice)
    float4 h4 = ((const float4*)(hs  + base))[lane];
    float4 b4 = ((const float4*)bias)[lane];
    float4 o;
    o.x = di * (a.x + h4.x) + b4.x;
    o.y = di * (a.y + h4.y) + b4.y;
    o.z = di * (a.z + h4.z) + b4.z;
    o.w = di * (a.w + h4.w) + b4.w;

    // wave32 reduction for row norm
    float ss = o.x * o.x + o.y * o.y + o.z * o.z + o.w * o.w;
    for (int off = 16; off >= 1; off >>= 1) ss += __shfl_xor(ss, off, 32);
    float nrm = fmaxf(sqrtf(ss), EPSV);
    float t = tanhf(nrm) / nrm;

    float4 z = make_float4(o.x * t, o.y * t, o.z * t, o.w * t);

    const size_t slice = (size_t)NNODES * FDIM;
    float4* o0 = (float4*)(out + base);                 // t=0 overwrites accumulator
    o0[lane] = o;
    ((float4*)(out + slice * 1 + base))[lane] = o;
    ((float4*)(out + slice * 2 + base))[lane] = o;
    ((float4*)(out + slice * 3 + base))[lane] = o;
    ((float4*)(out + slice * TREP + base))[lane] = z;   // z after the 4 repeats
}

extern "C" void kernel_launch(void* const* d_in, const int* in_sizes, int n_in,
                              void* d_out, int out_size, void* d_ws, size_t ws_size,
                              hipStream_t stream) {
    const float* x  = (const float*)d_in[0];
    const int*   ei = (const int*)d_in[1];       // (2, E): src then dst
    const float* W  = (const float*)d_in[2];
    const float* b  = (const float*)d_in[3];
    const int* src = ei;
    const int* dst = ei + NEDGES;
    float* out = (float*)d_out;

    float* deg = (float*)d_ws;                   // N floats (becomes dinv)
    float* hs  = deg + NNODES;                   // N*128 floats, 16B aligned

    deg_init_kernel <<<(NNODES + 255) / 256, 256, 0, stream>>>(deg);
    deg_accum_kernel<<<(NEDGES + 255) / 256, 256, 0, stream>>>(dst, deg);
    dinv_kernel     <<<(NNODES + 255) / 256, 256, 0, stream>>>(deg);
    zero_acc_kernel <<<(NNODES * FDIM / 4 + 255) / 256, 256, 0, stream>>>((float4*)out);
    gemm_scale_kernel<<<NNODES / 16, 256, 0, stream>>>(x, W, deg, hs);
    scatter_kernel  <<<(NEDGES + 7) / 8, 256, 0, stream>>>(src, dst, hs, out);
    finalize_kernel <<<(NNODES + 7) / 8, 256, 0, stream>>>(hs, deg, b, out);
}